// MultiHeadAttentionLayer_34351148433891
// MI455X (gfx1250) — compile-verified
//
#include <hip/hip_runtime.h>
#include <hip/hip_bf16.h>

typedef float v2f __attribute__((ext_vector_type(2)));
typedef float v4f __attribute__((ext_vector_type(4)));
typedef float v8f __attribute__((ext_vector_type(8)));

#define N_NODES 50000
#define N_EDGES 800000
#define IN_DIM  128
#define HD      128              // NUM_HEADS * OUT_DIM
#define NQ      (N_NODES * HD)   // floats per node-projection buffer
#define BPACK_FLOATS 16384       // 32 ks * 2 hi * 16 lo * 8 j * 2 = 64KB per W

// ---------------------------------------------------------------------------
// fp32 WMMA: D = A(16x4) * B(4x16) + C(16x16), exact fp32 accumulate.
// A frag (2 VGPR): lane lo=row M, hi half selects K = 2*hi .. 2*hi+1
// B frag (2 VGPR): lane lo=col N, hi half selects K = 2*hi .. 2*hi+1
// C/D  (8 VGPR):   component r -> row M = r + 8*hi, lane lo -> col N
// ---------------------------------------------------------------------------
__device__ __forceinline__ v8f wmma_f32(v2f a, v2f b, v8f c) {
    return __builtin_amdgcn_wmma_f32_16x16x4_f32(
        /*neg_a=*/false, a, /*neg_b=*/false, b,
        /*c_mod=*/(short)0, c, /*reuse_a=*/false, /*reuse_b=*/false);
}

// out_tile[16 x 128] = A_tile[16 x 128] @ W[128 x 128] + bias[128]
// Bpack: fragment-ready weights, indexed [((ks*2+hi)*16+lo)*16 + j*2 + c]
// so each lane's 8 B-fragments for one k-step are 64 contiguous bytes.
__device__ __forceinline__ void gemm_tile_16x128(const float* __restrict__ A,
                                                 const float* __restrict__ Bpack,
                                                 const float* __restrict__ bias,
                                                 int lo, int hi, v8f acc[8]) {
#pragma unroll
    for (int j = 0; j < 8; ++j) {
        float bb = bias[j * 16 + lo];
        acc[j] = (v8f){bb, bb, bb, bb, bb, bb, bb, bb};
    }
    for (int ks = 0; ks < 32; ++ks) {
        const v2f a = *(const v2f*)(A + lo * IN_DIM + 4 * ks + 2 * hi);
        const float* bp = Bpack + (size_t)((ks * 2 + hi) * 16 + lo) * 16;
        const v4f b01 = *(const v4f*)(bp + 0);   // j=0,1
        const v4f b23 = *(const v4f*)(bp + 4);   // j=2,3
        const v4f b45 = *(const v4f*)(bp + 8);   // j=4,5
        const v4f b67 = *(const v4f*)(bp + 12);  // j=6,7
        acc[0] = wmma_f32(a, (v2f){b01.x, b01.y}, acc[0]);
        acc[1] = wmma_f32(a, (v2f){b01.z, b01.w}, acc[1]);
        acc[2] = wmma_f32(a, (v2f){b23.x, b23.y}, acc[2]);
        acc[3] = wmma_f32(a, (v2f){b23.z, b23.w}, acc[3]);
        acc[4] = wmma_f32(a, (v2f){b45.x, b45.y}, acc[4]);
        acc[5] = wmma_f32(a, (v2f){b45.z, b45.w}, acc[5]);
        acc[6] = wmma_f32(a, (v2f){b67.x, b67.y}, acc[6]);
        acc[7] = wmma_f32(a, (v2f){b67.z, b67.w}, acc[7]);
    }
}

// ---------------------------------------------------------------------------
// Kernel P: repack the 4 weight matrices into fragment-ready layout.
//   Bpack[w][((ks*2+hi)*16+lo)*16 + j*2 + c] = W_w[(4ks+2hi+c)*128 + 16j+lo]
// ---------------------------------------------------------------------------
__global__ void prep_kernel(const float* __restrict__ Wq, const float* __restrict__ Wk,
                            const float* __restrict__ Wv, const float* __restrict__ We,
                            float* __restrict__ Bpack) {
    int idx = blockIdx.x * blockDim.x + threadIdx.x;
    if (idx >= 4 * BPACK_FLOATS) return;
    const int w   = idx / BPACK_FLOATS;
    const int rem = idx % BPACK_FLOATS;
    const int p = rem >> 4;          // (ks*2+hi)*16 + lo
    const int q = rem & 15;          // j*2 + c
    const int ks = p >> 5;
    const int hi = (p >> 4) & 1;
    const int lo = p & 15;
    const int j = q >> 1;
    const int c = q & 1;
    const int k = 4 * ks + 2 * hi + c;
    const int n = 16 * j + lo;
    const float* Ws[4] = {Wq, Wk, Wv, We};
    Bpack[idx] = Ws[w][(size_t)k * HD + n];
}

// ---------------------------------------------------------------------------
// Kernel 0: zero the atomic-accumulation scratch (wV + z, contiguous).
// ---------------------------------------------------------------------------
__global__ void zero_kernel(float* __restrict__ p, int n) {
    int i = blockIdx.x * blockDim.x + threadIdx.x;
    if (i < n) p[i] = 0.0f;
}

// ---------------------------------------------------------------------------
// Kernel 1: node projections Q,K,V = v @ W* + b*   (one wave per 16-node tile)
// ---------------------------------------------------------------------------
__global__ void node_proj_kernel(const float* __restrict__ v,
                                 const float* __restrict__ Bpack,
                                 const float* __restrict__ bq,
                                 const float* __restrict__ bk,
                                 const float* __restrict__ bv,
                                 float* __restrict__ Qo, float* __restrict__ Ko,
                                 float* __restrict__ Vo, int ntiles) {
    const int lane = threadIdx.x & 31;
    const int lo = lane & 15, hi = lane >> 4;
    const int tile = blockIdx.x * (blockDim.x >> 5) + (threadIdx.x >> 5);
    if (tile >= ntiles) return;                 // uniform per wave
    const float* A = v + (size_t)tile * 16 * IN_DIM;

    const float* bs[3] = {bq, bk, bv};
    float* outs[3] = {Qo, Ko, Vo};
#pragma unroll
    for (int w = 0; w < 3; ++w) {
        v8f acc[8];
        gemm_tile_16x128(A, Bpack + (size_t)w * BPACK_FLOATS, bs[w], lo, hi, acc);
#pragma unroll
        for (int j = 0; j < 8; ++j)
#pragma unroll
            for (int r = 0; r < 8; ++r)
                outs[w][(size_t)(tile * 16 + r + 8 * hi) * HD + j * 16 + lo] = acc[j][r];
    }
}

// ---------------------------------------------------------------------------
// Kernel 2: fused edge phase (one wave per 16-edge tile):
//   proj_e tile via WMMA; score = K[src]*Q[dst]*0.25*proj_e; write e_out;
//   per-head sum -> s = exp(clamp(.,-5,5)); atomic scatter of V[src]*env*s, s.
//   Head h == col-block j (OUT_DIM == 16): head reduction is a 16-lane
//   butterfly inside each half-wave.
// ---------------------------------------------------------------------------
__global__ void edge_kernel(const float* __restrict__ e,
                            const float* __restrict__ BpackE, const float* __restrict__ be,
                            const float* __restrict__ envelope,
                            const int* __restrict__ src, const int* __restrict__ dst,
                            const float* __restrict__ Qp, const float* __restrict__ Kp,
                            const float* __restrict__ Vp,
                            float* __restrict__ wV, float* __restrict__ z,
                            float* __restrict__ e_out, int ntiles) {
    const int lane = threadIdx.x & 31;
    const int lo = lane & 15, hi = lane >> 4;
    const int tile = blockIdx.x * (blockDim.x >> 5) + (threadIdx.x >> 5);
    if (tile >= ntiles) return;                 // uniform per wave

    // --- edge projection for this 16-edge tile ---
    v8f pe[8];
    gemm_tile_16x128(e + (size_t)tile * 16 * IN_DIM, BpackE, be, lo, hi, pe);

    // --- per-edge metadata (edge row m = r + 8*hi) ---
    int si[8], di[8];
    float env[8];
#pragma unroll
    for (int r = 0; r < 8; ++r) {
        const int eg = tile * 16 + r + 8 * hi;
        si[r] = src[eg];
        di[r] = dst[eg];
        env[r] = envelope[eg];
    }

#pragma unroll
    for (int j = 0; j < 8; ++j) {               // j == head index
        const int col = j * 16 + lo;            // col == h*16 + d
#pragma unroll
        for (int r = 0; r < 8; ++r) {
            const int eg = tile * 16 + r + 8 * hi;
            const float kv = Kp[(size_t)si[r] * HD + col];
            const float qv = Qp[(size_t)di[r] * HD + col];
            const float sc = kv * qv * 0.25f * pe[j][r];   // /sqrt(16)
            e_out[(size_t)eg * HD + col] = sc;

            // sum over d (16 lanes of this half-wave)
            float sum = sc;
            sum += __shfl_xor(sum, 1, 16);
            sum += __shfl_xor(sum, 2, 16);
            sum += __shfl_xor(sum, 4, 16);
            sum += __shfl_xor(sum, 8, 16);
            const float s = __expf(fminf(fmaxf(sum, -5.0f), 5.0f));

            const float vv = Vp[(size_t)si[r] * HD + col] * env[r];
            atomicAdd(&wV[(size_t)di[r] * HD + col], vv * s);
            if (lo == 0) atomicAdd(&z[(size_t)di[r] * 8 + j], s);
        }
    }
}

// ---------------------------------------------------------------------------
// Kernel 3: v_out = wV / (z + 1e-6)
// ---------------------------------------------------------------------------
__global__ void finalize_kernel(const float* __restrict__ wV,
                                const float* __restrict__ z,
                                float* __restrict__ v_out) {
    int i = blockIdx.x * blockDim.x + threadIdx.x;
    if (i < N_NODES * HD) {
        const int n = i >> 7;          // /128
        const int h = (i & 127) >> 4;  // head
        v_out[i] = wV[i] / (z[n * 8 + h] + 1e-6f);
    }
}

// ---------------------------------------------------------------------------
extern "C" void kernel_launch(void* const* d_in, const int* in_sizes, int n_in,
                              void* d_out, int out_size, void* d_ws, size_t ws_size,
                              hipStream_t stream) {
    const float* v        = (const float*)d_in[0];
    const float* e        = (const float*)d_in[1];
    const float* envelope = (const float*)d_in[2];
    const float* Wq       = (const float*)d_in[3];
    const float* bq       = (const float*)d_in[4];
    const float* Wk       = (const float*)d_in[5];
    const float* bk       = (const float*)d_in[6];
    const float* Wv       = (const float*)d_in[7];
    const float* bv       = (const float*)d_in[8];
    const float* We       = (const float*)d_in[9];
    const float* be       = (const float*)d_in[10];
    const int*   src      = (const int*)d_in[11];
    const int*   dst      = (const int*)d_in[12];

    float* ws    = (float*)d_ws;
    float* Qp    = ws;                   // [N,128]
    float* Kp    = ws + (size_t)NQ;      // [N,128]
    float* Vp    = ws + (size_t)2 * NQ;  // [N,128]
    float* wV    = ws + (size_t)3 * NQ;  // [N,128] atomic accum
    float* zp    = ws + (size_t)4 * NQ;  // [N,8]   atomic accum
    float* Bpack = zp + (size_t)N_NODES * 8;  // 4 x 16384 fragment-ready weights

    float* v_out = (float*)d_out;                        // [N,8,16]
    float* e_out = (float*)d_out + (size_t)N_NODES * HD; // [E,8,16]

    // P) repack weights into WMMA-fragment-ready layout (4 x 64KB)
    {
        const int n = 4 * BPACK_FLOATS;
        prep_kernel<<<(n + 255) / 256, 256, 0, stream>>>(Wq, Wk, Wv, We, Bpack);
    }
    // 0) zero atomic targets (wV and z are contiguous)
    {
        const int n = NQ + N_NODES * 8;
        zero_kernel<<<(n + 255) / 256, 256, 0, stream>>>(wV, n);
    }
    // 1) node projections: 3125 tiles of 16 nodes, 8 waves per block
    {
        const int ntiles = N_NODES / 16;                 // 3125
        const int blocks = (ntiles + 7) / 8;
        node_proj_kernel<<<blocks, 256, 0, stream>>>(v, Bpack, bq, bk, bv,
                                                     Qp, Kp, Vp, ntiles);
    }
    // 2) fused edge phase: 50000 tiles of 16 edges, 8 waves per block
    {
        const int ntiles = N_EDGES / 16;                 // 50000
        const int blocks = (ntiles + 7) / 8;
        edge_kernel<<<blocks, 256, 0, stream>>>(e, Bpack + (size_t)3 * BPACK_FLOATS, be,
                                                envelope, src, dst,
                                                Qp, Kp, Vp, wV, zp, e_out, ntiles);
    }
    // 3) finalize node output
    {
        const int n = N_NODES * HD;
        finalize_kernel<<<(n + 255) / 256, 256, 0, stream>>>(wV, zp, v_out);
    }
}